// TwoDimDDSPGenerator_23175643529773
// MI455X (gfx1250) — compile-verified
//
#include <hip/hip_runtime.h>
#include <cstddef>
#include <cstdint>

typedef __attribute__((ext_vector_type(2))) float v2f;
typedef __attribute__((ext_vector_type(8))) float v8f;

#define DEV_INLINE __device__ __forceinline__

DEV_INLINE float lrelu(float v){ return v >= 0.f ? v : 0.2f * v; }

// ---------------- CDNA5 async global->LDS staging helpers ----------------
// Preferred: clang builtin for GLOBAL_LOAD_ASYNC_TO_LDS_B32 (ASYNCcnt-tracked).
// Builtin signature (per hipcc diagnostic) takes int* params -> cast accordingly.
#if defined(__has_builtin)
#  if __has_builtin(__builtin_amdgcn_global_load_async_to_lds_b32)
#    define HAVE_ASYNC_LDS_BUILTIN 1
#  endif
#  if __has_builtin(__builtin_amdgcn_s_wait_asynccnt)
#    define HAVE_WAIT_ASYNC_BUILTIN 1
#  endif
#endif

DEV_INLINE void async_ld32(const float* g, float* l){
#if defined(HAVE_ASYNC_LDS_BUILTIN)
  __builtin_amdgcn_global_load_async_to_lds_b32(
      (__attribute__((address_space(1))) int*)(uintptr_t)g,
      (__attribute__((address_space(3))) int*)(uint32_t)(uintptr_t)l,
      0, 0);
#else
  uint64_t ga = (uint64_t)(uintptr_t)g;
  uint32_t la = (uint32_t)(uintptr_t)l;
  asm volatile("global_load_async_to_lds_b32 %0, %1, off"
               :: "v"(la), "v"(ga) : "memory");
#endif
}

DEV_INLINE void wait_async(){
#if defined(HAVE_WAIT_ASYNC_BUILTIN)
  __builtin_amdgcn_s_wait_asynccnt(0);
#else
  asm volatile("s_wait_asynccnt 0" ::: "memory");
#endif
  asm volatile("" ::: "memory");   // compiler barrier: LDS reads stay below the wait
}

// ======================= conv1: Ci=1, direct (only 27 MACs/out) =======================
__global__ void conv1_kernel(const float* __restrict__ x, const float* __restrict__ w,
                             const float* __restrict__ bias, float* __restrict__ out){
  int idx = blockIdx.x * blockDim.x + threadIdx.x;      // 2*32*256*512 = 8388608
  int wo = idx & 511; int h = (idx >> 9) & 255; int co = (idx >> 17) & 31; int b = idx >> 22;
  float acc = bias[co];
  for (int kh = 0; kh < 3; ++kh){
    int hi = h + kh - 1; if (hi < 0 || hi >= 256) continue;
    const float* row = x + ((size_t)b * 256 + hi) * 512;
    const float* wr  = w + (co * 3 + kh) * 9;
    for (int kw = 0; kw < 9; ++kw){
      int wi = wo + kw - 4; if (wi < 0 || wi >= 512) continue;
      acc += row[wi] * wr[kw];
    }
  }
  out[idx] = lrelu(acc);   // layout ((b*32+co)*256+h)*512+wo == idx
}

// ============ implicit-GEMM conv: LDS-staged patch + V_WMMA_F32_16X16X4_F32 ============
// One wave per 16(co) x 16(w) output tile at fixed (b, ho).
// GEMM K is permuted to (kh, kw, ci); CI % 4 == 0 means each WMMA K-group of 4 sits in
// one (kh,kw) -> no div/mod in the hot loop, B comes from LDS with zero-padded borders.
// A (16x4 f32): lane<16 -> m=lane, K=+0,+1 ; lane>=16 -> m=lane-16, K=+2,+3
// B (4x16 f32): lane<16 -> n=lane, rows +0,+1 ; lane>=16 -> n=lane-16, rows +2,+3
// C/D (16x16 f32): vgpr r -> m = r + (lane<16 ? 0 : 8), n = lane&15
template<int CI, int KH, int KW, int SH, int PH, int PW>
__global__ void __launch_bounds__(32) conv_wmma_kernel(
    const float* __restrict__ in, const float* __restrict__ wgt,
    const float* __restrict__ bias, float* __restrict__ out,
    int HI, int HO, int WO, int CO)
{
  constexpr int KHW = KH * KW;
  constexpr int K = CI * KHW;
  constexpr int WP = 16 + KW - 1;            // staged patch width
  static_assert((CI & 3) == 0, "CI must be divisible by 4");
  __shared__ float patch[CI * KH * WP];

  const int lane = threadIdx.x;
  const int wbase = blockIdx.x * 16;
  const int ho = blockIdx.y;
  const int cotiles = CO >> 4;
  const int b = blockIdx.z / cotiles;
  const int co_base = (blockIdx.z % cotiles) << 4;
  const int mn = lane & 15;                  // m for A-row, n for B-col
  const int kadd = (lane < 16) ? 0 : 2;
  const int WI = WO;                         // stride_w==1, pad keeps width for all uses
  const int wn = wbase + mn;

  // ---- stage input patch into LDS (async, zero-padded borders) ----
  const float* inb = in + (size_t)b * CI * HI * WI;
  for (int idx = lane; idx < CI * KH * WP; idx += 32){
    int row = idx / WP, col = idx - row * WP;
    int ci = row / KH, kh = row - ci * KH;
    int hi = ho * SH + kh - PH;
    int wi = wbase + col - PW;
    if (hi >= 0 && hi < HI && wi >= 0 && wi < WI)
      async_ld32(inb + ((size_t)ci * HI + hi) * WI + wi, &patch[idx]);
    else
      patch[idx] = 0.f;                      // DScnt-tracked; disjoint from async writes
  }
  wait_async();                              // single wave per block: no barrier needed

  // ---- GEMM: K = (kh, kw, ci-groups-of-4) ----
  const float* wrow = wgt + (size_t)(co_base + mn) * K;  // this lane's A row (OIHW flat)
  v8f acc = {0.f,0.f,0.f,0.f,0.f,0.f,0.f,0.f};
  for (int kh = 0; kh < KH; ++kh){
    for (int kw = 0; kw < KW; ++kw){
      const int rwk = kh * KW + kw;          // weight offset within one ci plane
      const int colb = mn + kw;              // B column in the patch
      for (int cib = 0; cib < CI / 4; ++cib){
        const int cia = cib * 4 + kadd;
        v2f a;  a.x  = wrow[cia * KHW + rwk];
                a.y  = wrow[cia * KHW + rwk + KHW];
        const int li = (cia * KH + kh) * WP + colb;
        v2f bv; bv.x = patch[li];
                bv.y = patch[li + KH * WP];
        acc = __builtin_amdgcn_wmma_f32_16x16x4_f32(false, a, false, bv, (short)0, acc,
                                                    false, false);
      }
    }
  }

  const int mtop = (lane < 16) ? 0 : 8;
  #pragma unroll
  for (int r2 = 0; r2 < 8; ++r2){
    int co = co_base + mtop + r2;
    float v = acc[r2] + bias[co];
    out[(((size_t)b * CO + co) * HO + ho) * WO + wn] = lrelu(v);
  }
}

// ======================= p-conv (Co=2) fused with loud/f knot generation =======================
__global__ void posc_kernel(const float* __restrict__ pre, const float* __restrict__ wp,
                            const float* __restrict__ bp, float* __restrict__ loudK,
                            float* __restrict__ fK){
  int idx = blockIdx.x * blockDim.x + threadIdx.x;      // 2*128*512 = 131072
  int wq = idx & 511; int h = (idx >> 9) & 127; int b = idx >> 16;
  float p0 = bp[0], p1 = bp[1];
  for (int ci = 0; ci < 32; ++ci){
    for (int kh = 0; kh < 3; ++kh){
      int hi = h + kh - 1; if (hi < 0 || hi >= 128) continue;
      const float* row = pre + (((size_t)b * 32 + ci) * 128 + hi) * 512;
      const float* w0 = wp + ((0 * 32 + ci) * 3 + kh) * 3;
      const float* w1 = wp + ((1 * 32 + ci) * 3 + kh) * 3;
      for (int kw = 0; kw < 3; ++kw){
        int wi = wq + kw - 1; if (wi < 0 || wi >= 512) continue;
        float v = row[wi];
        p0 += v * w0[kw];
        p1 += v * w1[kw];
      }
    }
  }
  loudK[idx] = p0 * p0;
  // geomspace(20, 5512.5, 128) band for oscillator h
  const float ratio = 275.625f;       // 5512.5/20
  float stop_h = 20.f * powf(ratio, (float)h * (1.f / 127.f));
  float prev   = (h == 0) ? 0.f : 20.f * powf(ratio, (float)(h - 1) * (1.f / 127.f));
  float sg = 1.f / (1.f + expf(-p1));
  fK[idx] = prev + sg * (stop_h - prev);  // clip happens post-upsample (matches reference)
}

// ======================= linear resize along W (half-pixel, edge-clamped) =======================
__global__ void resize_w_kernel(const float* __restrict__ in, float* __restrict__ out,
                                int rows, int win, int wout){
  int idx = blockIdx.x * blockDim.x + threadIdx.x;
  int total = rows * wout; if (idx >= total) return;
  int t = idx % wout; int r = idx / wout;
  float inx = ((float)t + 0.5f) * ((float)win / (float)wout) - 0.5f;
  int i0 = (int)floorf(inx);
  float fr = inx - (float)i0;
  int ia = min(max(i0, 0), win - 1);
  int ib = min(max(i0 + 1, 0), win - 1);
  const float* ri = in + (size_t)r * win;
  out[idx] = ri[ia] * (1.f - fr) + ri[ib] * fr;
}

// ======================= n3 conv (Co=17) fused with square -> n_l =======================
__global__ void n3_kernel(const float* __restrict__ n2r, const float* __restrict__ w,
                          const float* __restrict__ bias, float* __restrict__ nl){
  int idx = blockIdx.x * blockDim.x + threadIdx.x;      // 2*17*8192 = 278528
  if (idx >= 2 * 17 * 8192) return;
  int s = idx & 8191; int k = (idx >> 13) % 17; int b = idx / (17 * 8192);
  float acc = bias[k];
  for (int ci = 0; ci < 16; ++ci){
    for (int kh = 0; kh < 16; ++kh){
      const float* row = n2r + (((size_t)b * 16 + ci) * 16 + kh) * 8192;
      const float* wr  = w + ((k * 16 + ci) * 16 + kh) * 3;
      for (int kw = 0; kw < 3; ++kw){
        int wi = s + kw - 1; if (wi < 0 || wi >= 8192) continue;
        acc += row[wi] * wr[kw];
      }
    }
  }
  nl[((size_t)b * 17 + k) * 8192 + s] = acc * acc;
}

// ======================= harmonic branch: hierarchical phase scan =======================
DEV_INLINE float lerp512(const float* __restrict__ row, int t){
  float inx = ((float)t + 0.5f) * (1.f / 256.f) - 0.5f;  // 512 -> 131072 upsample
  int i0 = (int)floorf(inx);
  float fr = inx - (float)i0;
  int ia = min(max(i0, 0), 511);
  int ib = min(max(i0 + 1, 0), 511);
  return row[ia] * (1.f - fr) + row[ib] * fr;
}

#define DPHI_SCALE 5.6990343e-4f   // 2*pi/11025

__global__ void harm_chunksum_kernel(const float* __restrict__ fK, float* __restrict__ cbuf){
  __shared__ float sc[256];
  int tid = threadIdx.x;
  int c = blockIdx.x & 511; int o = (blockIdx.x >> 9) & 127; int b = blockIdx.x >> 16;
  int t = c * 256 + tid;
  const float* frow = fK + (size_t)(b * 128 + o) * 512;
  float fv = lerp512(frow, t);
  fv = fminf(fmaxf(fv, 20.f), 5512.5f);
  sc[tid] = fv * DPHI_SCALE;
  __syncthreads();
  for (int off = 128; off > 0; off >>= 1){
    if (tid < off) sc[tid] += sc[tid + off];
    __syncthreads();
  }
  if (tid == 0) cbuf[(size_t)(b * 128 + o) * 512 + c] = sc[0];
}

__global__ void harm_scan_kernel(float* __restrict__ cbuf){
  if (threadIdx.x != 0) return;                          // 256 rows, 512 entries each
  float* row = cbuf + (size_t)blockIdx.x * 512;
  float run = 0.f;
  for (int c = 0; c < 512; ++c){ float tmp = row[c]; row[c] = run; run += tmp; }
}

__global__ void harm_phase_kernel(const float* __restrict__ fK, const float* __restrict__ loudK,
                                  const float* __restrict__ cpref, float* __restrict__ harm){
  __shared__ float sc[256];
  int tid = threadIdx.x;
  int chunk = blockIdx.x;         // 0..511
  int b = blockIdx.y;             // 0..1
  int t = chunk * 256 + tid;
  float acc = 0.f;
  for (int o = 0; o < 128; ++o){
    size_t rowoff = (size_t)(b * 128 + o) * 512;
    float fv = lerp512(fK + rowoff, t);
    fv = fminf(fmaxf(fv, 20.f), 5512.5f);
    float dphi = fv * DPHI_SCALE;
    sc[tid] = dphi;
    __syncthreads();
    for (int off = 1; off < 256; off <<= 1){
      float add = (tid >= off) ? sc[tid - off] : 0.f;
      __syncthreads();
      sc[tid] += add;
      __syncthreads();
    }
    float phase = cpref[rowoff + chunk] + sc[tid];       // inclusive scan -> cumsum
    float lv = lerp512(loudK + rowoff, t);
    acc += sinf(phase) * lv;
  }
  harm[(size_t)b * 131072 + t] = acc;
}

// ======= noise synthesis: rfft*scale*irfft == 32-tap circular conv; one wave32/frame =======
__global__ void frames_kernel(const float* __restrict__ noise, const float* __restrict__ nl,
                              float* __restrict__ audio){
  __shared__ float xs[8][32];
  __shared__ float ks[8][32];
  int tid = threadIdx.x;
  int g = tid >> 5, j = tid & 31;
  int fs = blockIdx.x * 8 + g;          // 0..16383 (b*8192+s)
  int b = fs >> 13, s = fs & 8191;
  int idx = s * 16 + j;
  xs[g][j] = (idx < 131072) ? (noise[(size_t)b * 131072 + idx] * 2.f - 1.f) : 0.f;
  const float* sl = nl + (size_t)b * 17 * 8192 + s;
  float kd = sl[0];
  #pragma unroll
  for (int k = 1; k < 16; ++k)
    kd += 2.f * sl[(size_t)k * 8192] * cosf((float)(k * j) * 0.19634954f);  // pi/16
  kd += sl[(size_t)16 * 8192] * ((j & 1) ? -1.f : 1.f);
  ks[g][j] = kd * (1.f / 32.f);
  __syncthreads();
  float a = 0.f;
  #pragma unroll
  for (int jp = 0; jp < 32; ++jp)
    a += xs[g][jp] * ks[g][(j - jp) & 31];
  audio[(size_t)fs * 32 + j] = a;
}

// ======================= overlap-add + harmonic + crop =======================
__global__ void final_kernel(const float* __restrict__ harm, const float* __restrict__ audio,
                             float* __restrict__ out){
  int idx = blockIdx.x * blockDim.x + threadIdx.x;   // 2*122880 = 245760
  int t = idx % 122880; int b = idx / 122880;
  int u = t + 4096;
  const float* ab = audio + (size_t)b * 8192 * 32;
  float nb = ab[(size_t)(u >> 4) * 32 + (u & 15)];   // first-half contribution
  int v = u - 16;                                    // u >= 4096, always valid
  nb += ab[(size_t)(v >> 4) * 32 + 16 + (v & 15)];   // second-half contribution
  out[idx] = harm[(size_t)b * 131072 + u] + nb;
}

// ======================= host launcher =======================
extern "C" void kernel_launch(void* const* d_in, const int* in_sizes, int n_in,
                              void* d_out, int out_size, void* d_ws, size_t ws_size,
                              hipStream_t stream) {
  (void)in_sizes; (void)n_in; (void)out_size; (void)ws_size;
  const float* x     = (const float*)d_in[0];
  const float* noise = (const float*)d_in[1];
  const float* wm1 = (const float*)d_in[2];  const float* bm1 = (const float*)d_in[3];
  const float* wm2 = (const float*)d_in[4];  const float* bm2 = (const float*)d_in[5];
  const float* wm3 = (const float*)d_in[6];  const float* bm3 = (const float*)d_in[7];
  const float* wm4 = (const float*)d_in[8];  const float* bm4 = (const float*)d_in[9];
  const float* wp  = (const float*)d_in[10]; const float* bp  = (const float*)d_in[11];
  const float* wn1 = (const float*)d_in[12]; const float* bn1 = (const float*)d_in[13];
  const float* wn2 = (const float*)d_in[14]; const float* bn2 = (const float*)d_in[15];
  const float* wn3 = (const float*)d_in[16]; const float* bn3 = (const float*)d_in[17];
  float* out = (float*)d_out;
  float* ws  = (float*)d_ws;

  // workspace layout (floats). Peak requirement: 25,165,824 floats (~100.7 MB)
  const size_t F_H1    = 0;          // 8388608   (h1, later reused as h3)
  const size_t F_H2    = 8388608;    // 16777216  (h2, later reused: pre + everything below)
  const size_t F_H3    = 0;          // reuse of slot A
  const size_t F_PRE   = 8388608;    // 4194304
  const size_t F_LOUDK = 12582912;   // 131072
  const size_t F_FK    = 12713984;   // 131072
  const size_t F_CPREF = 12845056;   // 131072
  const size_t F_HARM  = 12976128;   // 262144
  const size_t F_N1    = 13238272;   // 524288
  const size_t F_N1R   = 13762560;   // 2097152
  const size_t F_N2    = 15859712;   // 1048576
  const size_t F_N2R   = 16908288;   // 4194304
  const size_t F_NL    = 21102592;   // 278528
  const size_t F_AUDIO = 21381120;   // 524288

  // ---- melody trunk ----
  conv1_kernel<<<32768, 256, 0, stream>>>(x, wm1, bm1, ws + F_H1);

  // conv2: (2,32,256,512) -> (2,64,256,512)
  conv_wmma_kernel<32,3,9,1,1,4><<<dim3(32, 256, 8), 32, 0, stream>>>(
      ws + F_H1, wm2, bm2, ws + F_H2, 256, 256, 512, 64);
  // conv3: stride (2,1) -> (2,64,128,512)
  conv_wmma_kernel<64,3,9,2,1,4><<<dim3(32, 128, 8), 32, 0, stream>>>(
      ws + F_H2, wm3, bm3, ws + F_H3, 256, 128, 512, 64);
  // conv4: -> (2,32,128,512) == pre
  conv_wmma_kernel<64,3,9,1,1,4><<<dim3(32, 128, 4), 32, 0, stream>>>(
      ws + F_H3, wm4, bm4, ws + F_PRE, 128, 128, 512, 32);

  // p-conv + oscillator knots
  posc_kernel<<<512, 256, 0, stream>>>(ws + F_PRE, wp, bp, ws + F_LOUDK, ws + F_FK);

  // ---- noise branch ----
  // n1: (2,32,128,512) -> (2,16,32,512), k=(9,3), stride(4,1), pad(4,1)
  conv_wmma_kernel<32,9,3,4,4,1><<<dim3(32, 32, 2), 32, 0, stream>>>(
      ws + F_PRE, wn1, bn1, ws + F_N1, 128, 32, 512, 16);
  // resize W 512 -> 2048 (rows = 2*16*32)
  resize_w_kernel<<<8192, 256, 0, stream>>>(ws + F_N1, ws + F_N1R, 1024, 512, 2048);
  // n2: (2,16,32,2048) -> (2,16,16,2048), k=(5,3), stride(2,1), pad(2,1)
  conv_wmma_kernel<16,5,3,2,2,1><<<dim3(128, 16, 2), 32, 0, stream>>>(
      ws + F_N1R, wn2, bn2, ws + F_N2, 32, 16, 2048, 16);
  // resize W 2048 -> 8192 (rows = 2*16*16)
  resize_w_kernel<<<16384, 256, 0, stream>>>(ws + F_N2, ws + F_N2R, 512, 2048, 8192);
  // n3 + square -> n_l (2,17,8192)
  n3_kernel<<<1088, 256, 0, stream>>>(ws + F_N2R, wn3, bn3, ws + F_NL);

  // ---- harmonic synthesis (hierarchical cumsum) ----
  harm_chunksum_kernel<<<131072, 256, 0, stream>>>(ws + F_FK, ws + F_CPREF);
  harm_scan_kernel<<<256, 32, 0, stream>>>(ws + F_CPREF);
  harm_phase_kernel<<<dim3(512, 2), 256, 0, stream>>>(ws + F_FK, ws + F_LOUDK,
                                                      ws + F_CPREF, ws + F_HARM);

  // ---- noise synthesis (per-frame circular conv) ----
  frames_kernel<<<2048, 256, 0, stream>>>(noise, ws + F_NL, ws + F_AUDIO);

  // ---- overlap-add + mix + crop ----
  final_kernel<<<960, 256, 0, stream>>>(ws + F_HARM, ws + F_AUDIO, out);
}